// OctahedralRealSHT_3770981286260
// MI455X (gfx1250) — compile-verified
//
#include <hip/hip_runtime.h>

// ---------------------------------------------------------------------------
// Octahedral real SHT for MI455X (gfx1250, wave32, WMMA).
//
// Stage 1: per-ring real DFT as WMMA f32 16x16x4 GEMM with on-the-fly
//          twiddles -> X[m][bc][k] (real & imag planes in d_ws).
//          Twiddle angles via exact integer phase recurrence (no runtime %,
//          no guarded loads: every nlon is a multiple of 4).
// Stage 2: per-m Legendre GEMM out[bc,l,m] = sum_k X[bc,k,m] * w[m,l,k],
//          WMMA f32 16x16x4, K=256 fully register-accumulated.
//
// Both WMMA operands stream from global with contiguous b64 loads thanks to
// the [m][bc][k] intermediate layout + w's native [m][l][k] layout.
// ---------------------------------------------------------------------------

typedef __attribute__((ext_vector_type(2))) float v2f;
typedef __attribute__((ext_vector_type(8))) float v8f;

#define NLAT    256
#define LMAX    256
#define MMAX    256
#define BC      512          // BATCH * CH
#define NPOINTS 70144

__device__ __forceinline__ v8f wmma_f32x4(v2f a, v2f b, v8f c) {
  // V_WMMA_F32_16X16X4_F32: D = A(16x4) x B(4x16) + C(16x16), fp32 native.
  return __builtin_amdgcn_wmma_f32_16x16x4_f32(
      /*neg_a=*/false, a, /*neg_b=*/false, b,
      /*c_mod=*/(short)0, c, /*reuse_a=*/false, /*reuse_b=*/false);
}

// Ring geometry (closed forms; no host tables needed).
__device__ __forceinline__ int ring_nlon(int k) {
  int i = (k < 128) ? k : 255 - k;
  return 4 * (i + 1) + 16;                       // 20 .. 528, always %4 == 0
}
__device__ __forceinline__ int ring_slon(int k) {
  if (k <= 128) return 2 * k * k + 18 * k;       // sum of (4i+20), i<k
  return 35072 + (k - 128) * (786 - 2 * k);      // mirrored upper half
}

// ---------------------------------------------------------------------------
// Stage 1: ring DFT.  grid = (ring=256, mtile=16, bcgrp=4), block = 256 (8 waves)
// Wave w handles bc-tile = blockIdx.z*8 + w.
// Output layout: X[m][bc][k]  (k contiguous -> stage-2 A rows are b64 loads)
// ---------------------------------------------------------------------------
__global__ __launch_bounds__(256)
void sht_stage1_dft(const float* __restrict__ x,
                    float* __restrict__ Xr, float* __restrict__ Xi)
{
  const int wave = threadIdx.x >> 5;
  const int lane = threadIdx.x & 31;
  const int half = lane >> 4;       // 0: K={0,1}, 1: K={2,3}
  const int lm   = lane & 15;

  const int k    = blockIdx.x;                       // latitude ring
  const int m0   = blockIdx.y << 4;                  // m tile base
  const int bc0  = ((blockIdx.z << 3) + wave) << 4;  // bc tile base

  const int nlon = ring_nlon(k);
  const int slon = ring_slon(k);
  const int nf   = (nlon >> 1) + 1;
  const int mlim = nf < MMAX ? nf : MMAX;

  const int   m      = m0 + lm;                      // this lane's Fourier mode
  const float twopi  = 6.28318530717958647692f;
  const float invn   = 1.0f / (float)nlon;
  const float angsc  = -twopi * invn;

  // Exact integer phase recurrence: r_j = (m*j) mod nlon, advanced by
  // step = (4m) mod nlon each K-iteration.  The only runtime mods are here,
  // hoisted out of the loop; inside it's add + conditional subtract.
  const int mmod = m % nlon;
  const int step = (4 * m) % nlon;
  int r0 = (m * (half << 1)) % nlon;                 // phase at jj   = 2*half
  int r1 = r0 + mmod; if (r1 >= nlon) r1 -= nlon;    // phase at jj+1

  // A row for this lane: x[bc0+lm][slon + j]; 8B aligned (slon even, jj even).
  const float* __restrict__ p =
      x + (size_t)(bc0 + lm) * NPOINTS + slon + (half << 1);

  v8f cr = {};   // cos accumulation (real part)
  v8f ci = {};   // -sin accumulation (imag part)

  for (int j0 = 0; j0 < nlon; j0 += 4, p += 4) {
    const v2f a = *(const v2f*)p;                    // unconditional b64 load

    const float a0 = angsc * (float)r0;
    const float a1 = angsc * (float)r1;
    v2f bcos; bcos.x = __cosf(a0); bcos.y = __cosf(a1);
    v2f bsin; bsin.x = __sinf(a0); bsin.y = __sinf(a1);  // sin of neg angle

    cr = wmma_f32x4(a, bcos, cr);
    ci = wmma_f32x4(a, bsin, ci);

    r0 += step; if (r0 >= nlon) r0 -= nlon;
    r1 += step; if (r1 >= nlon) r1 -= nlon;
  }

  // X = 2*pi * fft / nlon ; zero-pad modes beyond nf (rfft length).
  const float scale = (m < mlim) ? (twopi * invn) : 0.0f;

  // C/D layout: VGPR i -> bc row (i, lanes 0-15) / (i+8, lanes 16-31); col = m.
#pragma unroll
  for (int i = 0; i < 8; ++i) {
    const int bcr = bc0 + i + (half << 3);
    const size_t idx = ((size_t)m * BC + bcr) * NLAT + k;
    Xr[idx] = cr[i] * scale;
    Xi[idx] = ci[i] * scale;
  }
}

// ---------------------------------------------------------------------------
// Stage 2: per-m Legendre GEMM.
// grid = (m=256, ltile=16, bcgrp=4), block = 256 (8 waves).
// out[bc,l,m] = sum_k X[m][bc][k] * w[m][l][k]; complex64 interleaved.
// ---------------------------------------------------------------------------
__global__ __launch_bounds__(256)
void sht_stage2_leg(const float* __restrict__ Xr, const float* __restrict__ Xi,
                    const float* __restrict__ w, float* __restrict__ out)
{
  const int wave = threadIdx.x >> 5;
  const int lane = threadIdx.x & 31;
  const int half = lane >> 4;
  const int lm   = lane & 15;

  const int mm  = blockIdx.x;                        // Fourier mode
  const int l0  = blockIdx.y << 4;                   // l tile base
  const int bc0 = ((blockIdx.z << 3) + wave) << 4;   // bc tile base

  // Per-lane contiguous K rows: A from X[m][bc][*], B from w[m][l][*] (both
  // already K-contiguous -> pure global_load_b64 streams, no LDS transpose).
  const float* __restrict__ xrRow = Xr + ((size_t)mm * BC + bc0 + lm) * NLAT;
  const float* __restrict__ xiRow = Xi + ((size_t)mm * BC + bc0 + lm) * NLAT;
  const float* __restrict__ wRow  = w  + ((size_t)mm * LMAX + l0 + lm) * NLAT;

  v8f cr = {};
  v8f ci = {};

#pragma unroll 4
  for (int k0 = 0; k0 < NLAT; k0 += 4) {
    const int jj = k0 + (half << 1);                 // 8B-aligned (jj even)
    const v2f ar = *(const v2f*)(xrRow + jj);
    const v2f ai = *(const v2f*)(xiRow + jj);
    const v2f bw = *(const v2f*)(wRow  + jj);
    cr = wmma_f32x4(ar, bw, cr);
    ci = wmma_f32x4(ai, bw, ci);
  }

  // Write complex64: (re, im) pairs, index [bc][l][m].
  const int l = l0 + lm;
#pragma unroll
  for (int i = 0; i < 8; ++i) {
    const int bcr = bc0 + i + (half << 3);
    const size_t cidx = ((size_t)bcr * LMAX + l) * MMAX + mm;
    v2f val; val.x = cr[i]; val.y = ci[i];
    *(v2f*)(out + 2 * cidx) = val;
  }
}

// ---------------------------------------------------------------------------
extern "C" void kernel_launch(void* const* d_in, const int* in_sizes, int n_in,
                              void* d_out, int out_size, void* d_ws, size_t ws_size,
                              hipStream_t stream) {
  const float* x = (const float*)d_in[0];   // (4,128,70144) f32
  const float* w = (const float*)d_in[1];   // (256,256,256) f32
  float* out = (float*)d_out;               // (4,128,256,256) complex64 interleaved

  // Workspace: Xr | Xi, each MMAX*BC*NLAT floats (134 MB each).
  float* Xr = (float*)d_ws;
  float* Xi = Xr + (size_t)MMAX * BC * NLAT;

  dim3 block(256);
  dim3 g1(NLAT, MMAX / 16, BC / (16 * 8));  // (256,16,4)
  sht_stage1_dft<<<g1, block, 0, stream>>>(x, Xr, Xi);

  dim3 g2(MMAX, LMAX / 16, BC / (16 * 8));  // (256,16,4)
  sht_stage2_leg<<<g2, block, 0, stream>>>(Xr, Xi, w, out);

  (void)in_sizes; (void)n_in; (void)out_size; (void)ws_size;
}